// OptBlockv2_45466523795646
// MI455X (gfx1250) — compile-verified
//
#include <hip/hip_runtime.h>
#include <stdint.h>

// Problem dims: S shops (M), NB bundles (K), SC styles*colors (N)
constexpr int S  = 2048;   // M
constexpr int NB = 1024;   // K
constexpr int SC = 4096;   // N

// Tiling
constexpr int BM = 128;
constexpr int BN = 256;
constexpr int BK = 32;
constexpr int ASTRIDE = 36;    // A LDS row stride (floats): 144B rows, 16B aligned, conflict-free b64 frags
constexpr int BSTRIDE = 264;   // B LDS row stride: 256 data + 8 pad dwords (2*264 % 64 == 16 -> halves disjoint)
constexpr int NCHUNK  = NB / BK;   // 32 (even)

typedef float v2f __attribute__((ext_vector_type(2)));
typedef float v8f __attribute__((ext_vector_type(8)));
typedef unsigned int u32x4 __attribute__((ext_vector_type(4)));
typedef int i32x4 __attribute__((ext_vector_type(4)));
typedef int i32x8 __attribute__((ext_vector_type(8)));

__device__ __forceinline__ float scale_p(float p, float a, float l) {
    bool over = ((a - p) < 0.0f && p > 0.0f) || ((l + p) < 0.0f && p < 0.0f);
    return p * (over ? 0.01f : 1.0f);
}

// Stage the 128x32 A tile (constraint multiplier fused) into LDS via VGPRs.
__device__ __forceinline__ void stage_A(float* __restrict__ Asb,
                                        const float* __restrict__ proba,
                                        const float* __restrict__ avail,
                                        const float* __restrict__ alloc,
                                        int bm0, int k0, int tid) {
    #pragma unroll
    for (int i = 0; i < 4; ++i) {
        int idx = tid + i * 256;          // 1024 float4 slots
        int row = idx >> 3;
        int c4  = idx & 7;
        int gm  = bm0 + row;
        int gk  = k0 + c4 * 4;
        const float4 pv = *(const float4*)(proba + (size_t)gm * NB + gk);
        const float4 av = *(const float4*)(avail + gk);
        const float4 lv = *(const float4*)(alloc + gk);
        float4 r;
        r.x = scale_p(pv.x, av.x, lv.x);
        r.y = scale_p(pv.y, av.y, lv.y);
        r.z = scale_p(pv.z, av.z, lv.z);
        r.w = scale_p(pv.w, av.w, lv.w);
        *(float4*)(&Asb[row * ASTRIDE + c4 * 4]) = r;
    }
}

// Issue one Tensor Data Mover DMA: 32x256 f32 tile of bun_sc -> LDS (rows padded
// 256 data dwords + 8 pad dwords = BSTRIDE). Called by wave 0 only; TENSORcnt tracked.
__device__ __forceinline__ void tdm_load_B(float* Bsb, const float* bun, int k0, int bn0) {
    uint64_t ga  = (uint64_t)(uintptr_t)(bun + (size_t)k0 * SC + bn0);
    uint32_t lds = (uint32_t)(uintptr_t)Bsb;   // generic LDS ptr low 32 bits = LDS byte offset

    u32x4 g0;
    g0[0] = 1u;                                          // count=1, user descriptor
    g0[1] = lds;                                         // lds_addr (bytes)
    g0[2] = (uint32_t)ga;                                // global_addr[31:0] (tile start)
    g0[3] = ((uint32_t)(ga >> 32) & 0x01FFFFFFu)         // global_addr[56:32]
          | (2u << 30);                                  // type=2 ("image")

    i32x8 g1;
    g1[0] = (int)((2u << 16)                             // data_size = 4 bytes
                | (1u << 20)                             // pad_enable
                | (7u << 22)                             // pad_interval = 256 dwords
                | (7u << 25));                           // pad_amount   = 8 dwords
    g1[1] = (int)((uint32_t)SC << 16);                   // tensor_dim0[15:0] -> bits[63:48]
    g1[2] = (int)(((uint32_t)SC >> 16)                   // tensor_dim0[31:16]
                | ((uint32_t)(NB & 0xFFFF) << 16));      // tensor_dim1[15:0]
    g1[3] = (int)(((uint32_t)NB >> 16)                   // tensor_dim1[31:16]
                | ((uint32_t)BN << 16));                 // tile_dim0 = 256
    g1[4] = (int)BK;                                     // tile_dim1 = 32, tile_dim2 = 0
    g1[5] = (int)SC;                                     // tensor_dim0_stride[31:0] (elements)
    g1[6] = 0;                                           // stride hi / dim1_stride lo
    g1[7] = 0;

    i32x4 z4 = {0, 0, 0, 0};                             // groups 2/3 unused (2D tensor)
    i32x8 z8 = {0, 0, 0, 0, 0, 0, 0, 0};
    __builtin_amdgcn_tensor_load_to_lds(g0, g1, z4, z4, z8, 0);
}

// 8 K-steps x 16 WMMA on one staged chunk.
__device__ __forceinline__ void compute_chunk(const float* __restrict__ Asb,
                                              const float* __restrict__ Bsb,
                                              v8f (&acc)[4][4],
                                              int mrow, int ncol, int kb) {
    #pragma unroll
    for (int kk = 0; kk < BK; kk += 4) {
        v2f a[4], b[4];
        #pragma unroll
        for (int mt = 0; mt < 4; ++mt)
            a[mt] = *(const v2f*)(&Asb[(mrow + mt * 16) * ASTRIDE + kk + kb]);
        #pragma unroll
        for (int nt = 0; nt < 4; ++nt) {
            v2f t;
            t.x = Bsb[(kk + kb    ) * BSTRIDE + ncol + nt * 16];
            t.y = Bsb[(kk + kb + 1) * BSTRIDE + ncol + nt * 16];
            b[nt] = t;
        }
        #pragma unroll
        for (int mt = 0; mt < 4; ++mt)
            #pragma unroll
            for (int nt = 0; nt < 4; ++nt)
                acc[mt][nt] = __builtin_amdgcn_wmma_f32_16x16x4_f32(
                    false, a[mt], false, b[nt], (short)0, acc[mt][nt], false, false);
    }
}

__global__ __launch_bounds__(256, 1)
void bundle_gemm_wmma(const float* __restrict__ proba,
                      const float* __restrict__ cur,
                      const float* __restrict__ bun,
                      const float* __restrict__ avail,
                      const float* __restrict__ alloc,
                      float* __restrict__ out)
{
    // Double-buffered tiles: 2*(18432 + 33792) = 104448 B of the 320 KB WGP LDS.
    __shared__ float As0[BM * ASTRIDE];
    __shared__ float As1[BM * ASTRIDE];
    __shared__ float Bs0[BK * BSTRIDE];
    __shared__ float Bs1[BK * BSTRIDE];

    const int tid  = threadIdx.x;
    const int lane = tid & 31;
    const int wave = tid >> 5;            // 8 waves, 2 (M) x 4 (N)
    const int wm0  = (wave >> 2) * 64;
    const int wn0  = (wave & 3) * 64;
    const int bm0  = blockIdx.y * BM;
    const int bn0  = blockIdx.x * BN;

    v8f acc[4][4];
    #pragma unroll
    for (int mt = 0; mt < 4; ++mt)
        #pragma unroll
        for (int nt = 0; nt < 4; ++nt)
            #pragma unroll
            for (int r = 0; r < 8; ++r)
                acc[mt][nt][r] = 0.0f;

    // WMMA f32 16x16x4 fragment addressing (ISA 7.12.2):
    //   lanes 0-15: M/N = lane, K = {0,1};  lanes 16-31: M/N = lane-16, K = {2,3}
    const int kb   = (lane >> 4) << 1;
    const int mrow = wm0 + (lane & 15);
    const int ncol = wn0 + (lane & 15);

    // Prologue: fill buffer 0.
    stage_A(As0, proba, avail, alloc, bm0, 0, tid);
    if (wave == 0) {
        tdm_load_B(Bs0, bun, 0, bn0);
        __builtin_amdgcn_s_wait_tensorcnt(0);
    }
    __syncthreads();

    // Software-pipelined chunk loop, unrolled x2 over named buffers so the
    // scheduler can overlap next-chunk staging (VGPR A path + TDM B path)
    // with the WMMA stream of the current chunk.
    for (int c = 0; c < NCHUNK; c += 2) {
        const int k0 = c * BK;

        // even chunk: compute buf0, stage chunk c+1 into buf1
        stage_A(As1, proba, avail, alloc, bm0, k0 + BK, tid);
        if (wave == 0) tdm_load_B(Bs1, bun, k0 + BK, bn0);
        compute_chunk(As0, Bs0, acc, mrow, ncol, kb);
        if (wave == 0) __builtin_amdgcn_s_wait_tensorcnt(0);
        __syncthreads();

        // odd chunk: compute buf1, stage chunk c+2 into buf0 (if any)
        if (c + 2 < NCHUNK) {
            stage_A(As0, proba, avail, alloc, bm0, k0 + 2 * BK, tid);
            if (wave == 0) tdm_load_B(Bs0, bun, k0 + 2 * BK, bn0);
        }
        compute_chunk(As1, Bs1, acc, mrow, ncol, kb);
        if (wave == 0) __builtin_amdgcn_s_wait_tensorcnt(0);
        __syncthreads();
    }

    // Epilogue: out = cur_inven + acc  (16x16 f32 C/D layout: VGPR r holds
    // row r + 8*(lane>=16), column lane&15).
    const int rowOff = (lane >> 4) * 8;
    #pragma unroll
    for (int mt = 0; mt < 4; ++mt) {
        #pragma unroll
        for (int nt = 0; nt < 4; ++nt) {
            int gm0 = bm0 + wm0 + mt * 16 + rowOff;
            int gn  = bn0 + wn0 + nt * 16 + (lane & 15);
            #pragma unroll
            for (int r = 0; r < 8; ++r) {
                size_t off = (size_t)(gm0 + r) * SC + gn;
                out[off] = cur[off] + acc[mt][nt][r];
            }
        }
    }
}

extern "C" void kernel_launch(void* const* d_in, const int* in_sizes, int n_in,
                              void* d_out, int out_size, void* d_ws, size_t ws_size,
                              hipStream_t stream) {
    (void)in_sizes; (void)n_in; (void)out_size; (void)d_ws; (void)ws_size;
    const float* proba = (const float*)d_in[0];  // [S*NB]
    const float* cur   = (const float*)d_in[1];  // [S, SC]
    const float* bun   = (const float*)d_in[2];  // [NB, SC]
    const float* avail = (const float*)d_in[3];  // [NB]
    const float* alloc = (const float*)d_in[4];  // [NB]
    float* out = (float*)d_out;                  // [S*SC]

    dim3 grid(SC / BN, S / BM);                  // 16 x 16 = 256 workgroups
    bundle_gemm_wmma<<<grid, 256, 0, stream>>>(proba, cur, bun, avail, alloc, out);
}